// GinDecoderLayer_68461778698669
// MI455X (gfx1250) — compile-verified
//
#include <hip/hip_runtime.h>

// Problem constants (fixed shapes from the reference)
#define BN 16      // batches
#define NN 8192    // nodes per batch
#define DD 256     // node feature dim
#define GG 16      // graphs per pack (segments)

#define THREADS 256
#define WAVES   (THREADS / 32)
#define NODES_PER_WG 256
#define WGS_PER_BATCH (NN / NODES_PER_WG)   // 32

typedef float v2f __attribute__((ext_vector_type(2)));
typedef float v8f __attribute__((ext_vector_type(8)));

// ---------------------------------------------------------------------------
// Zero the global accumulators (sums + counts) in workspace.
// ---------------------------------------------------------------------------
__global__ void gin_init_kernel(float* ws) {
    int t = blockIdx.x * blockDim.x + threadIdx.x;
    if (t < 2 * BN * GG) ws[t] = 0.0f;
}

// ---------------------------------------------------------------------------
// Main kernel: stream nodes once, per-row dot with W via V_WMMA_F32_16X16X4_F32,
// segment-accumulate scalars in LDS, one global atomic per (WG, g).
// ---------------------------------------------------------------------------
__global__ __launch_bounds__(THREADS) void gin_main_kernel(
    const float* __restrict__ nodes,   // [BN, NN, DD]
    const int*   __restrict__ nidx,    // [BN, NN]
    const float* __restrict__ W,       // [DD]
    float* __restrict__ gsum,          // [BN*GG]
    float* __restrict__ gcnt)          // [BN*GG]
{
    __shared__ __align__(16) float Wl[DD];          // W staged in LDS
    __shared__ float accum[GG];                     // per-WG segment sums
    __shared__ float cnts[GG];                      // per-WG segment counts
    __shared__ __align__(16) float stage[WAVES*16]; // per-wave dot staging

    const int tid  = threadIdx.x;
    const int lane = tid & 31;
    const int wave = tid >> 5;
    const int b    = blockIdx.x / WGS_PER_BATCH;
    const int slab = (blockIdx.x % WGS_PER_BATCH) * NODES_PER_WG;

    Wl[tid] = W[tid];                 // DD == THREADS == 256
    if (tid < GG)            accum[tid]      = 0.0f;
    else if (tid < 2 * GG)   cnts[tid - GG]  = 0.0f;
    __syncthreads();

    // A-matrix layout for V_WMMA_F32_16X16X4_F32 (16x4 f32):
    //   lanes 0-15 : row M=lane,     vgpr0=K0, vgpr1=K1
    //   lanes16-31 : row M=lane-16,  vgpr0=K2, vgpr1=K3
    const int mrow = lane & 15;
    const int koff = (lane >> 4) << 1;     // 0 or 2

    const float* baseN = nodes + (size_t)b * NN * DD;
    const int*   bidx  = nidx  + (size_t)b * NN;

    for (int chunk = slab + wave * 16; chunk < slab + NODES_PER_WG;
         chunk += WAVES * 16) {

        const float* arow = baseN + (size_t)(chunk + mrow) * DD + koff;

        v8f c = {0.f, 0.f, 0.f, 0.f, 0.f, 0.f, 0.f, 0.f};

        // D = A(16 nodes x 4 dims) * B(4 dims x 16 cols, all cols = W slice) + C
        // 64 K-steps cover the full 256-dim row; C column j == dot for all j.
        #pragma unroll 16
        for (int k = 0; k < DD; k += 4) {
            v2f a  = *(const v2f*)(arow + k);          // global_load_b64, coalesced
            v2f bv = *(const v2f*)(&Wl[k + koff]);     // ds_load_b64 (broadcast)
            c = __builtin_amdgcn_wmma_f32_16x16x4_f32(
                    /*neg_a=*/false, a, /*neg_b=*/false, bv,
                    /*c_mod=*/(short)0, c, /*reuse_a=*/false, /*reuse_b=*/false);
        }

        // C layout: vgpr r, lanes0-15 -> row M=r ; lanes16-31 -> row M=r+8.
        // Every column is identical, so lane 0 holds dots for nodes 0..7 and
        // lane 16 holds dots for nodes 8..15. Stage them to LDS.
        if ((lane & 15) == 0) {
            float* st = &stage[wave * 16 + (lane >> 4) * 8];
            ((float4*)st)[0] = make_float4(c[0], c[1], c[2], c[3]); // ds_store_b128
            ((float4*)st)[1] = make_float4(c[4], c[5], c[6], c[7]); // ds_store_b128
        }
        asm volatile("s_wait_dscnt 0x0" ::: "memory");   // CDNA5 split DS counter

        // Segment scatter: 16 scalar dots -> per-group LDS accumulators.
        if (lane < 16) {
            float dot = stage[wave * 16 + lane];
            int   g   = bidx[chunk + lane];              // 0..15
            atomicAdd(&accum[g], dot);                   // ds_add_f32
            atomicAdd(&cnts[g], 1.0f);                   // ds_add_f32
        }
    }
    __syncthreads();

    if (tid < GG) {
        atomicAdd(&gsum[b * GG + tid], accum[tid]);      // global_atomic_add_f32
        atomicAdd(&gcnt[b * GG + tid], cnts[tid]);
    }
}

// ---------------------------------------------------------------------------
// Finalize: out[b,g] = sum / max(cnt,1) + bias
// ---------------------------------------------------------------------------
__global__ void gin_final_kernel(const float* __restrict__ gsum,
                                 const float* __restrict__ gcnt,
                                 const float* __restrict__ bias,
                                 float* __restrict__ out) {
    int t = blockIdx.x * blockDim.x + threadIdx.x;
    if (t < BN * GG) {
        float cden = gcnt[t];
        if (cden < 1.0f) cden = 1.0f;
        out[t] = gsum[t] / cden + bias[0];
    }
}

// ---------------------------------------------------------------------------
// setup_inputs() order:
//  0 nodes [16,8192,256] f32   5 node_graph_idx [16,8192] i32
//  1 edges (unused)            6 edge_graph_idx (unused)
//  2 receivers (unused)        7 W [256,1] f32
//  3 senders (unused)          8 b [1] f32
//  4 global_latent (unused)    9 n_graphs_per_pack (scalar, ==16)
// ---------------------------------------------------------------------------
extern "C" void kernel_launch(void* const* d_in, const int* in_sizes, int n_in,
                              void* d_out, int out_size, void* d_ws, size_t ws_size,
                              hipStream_t stream) {
    (void)in_sizes; (void)n_in; (void)out_size; (void)ws_size;

    const float* nodes = (const float*)d_in[0];
    const int*   nidx  = (const int*)  d_in[5];
    const float* W     = (const float*)d_in[7];
    const float* bias  = (const float*)d_in[8];
    float*       out   = (float*)d_out;

    float* gsum = (float*)d_ws;          // BN*GG floats
    float* gcnt = gsum + BN * GG;        // BN*GG floats

    gin_init_kernel<<<1, 2 * BN * GG, 0, stream>>>((float*)d_ws);
    gin_main_kernel<<<BN * WGS_PER_BATCH, THREADS, 0, stream>>>(
        nodes, nidx, W, gsum, gcnt);
    gin_final_kernel<<<1, BN * GG, 0, stream>>>(gsum, gcnt, bias, out);
}